// RPN_24764781429426
// MI455X (gfx1250) — compile-verified
//
#include <hip/hip_runtime.h>
#include <hip/hip_bf16.h>
#include <math.h>

typedef __attribute__((ext_vector_type(2))) float v2f;
typedef __attribute__((ext_vector_type(8))) float v8f;
typedef __attribute__((ext_vector_type(4))) unsigned int v4u;
typedef __attribute__((ext_vector_type(8))) int v8i;
typedef __attribute__((ext_vector_type(4))) int v4i;

#define FH 50
#define FW 50
#define HW 2500
#define CIN 512
#define COUT 512
#define NB 8
#define NANCH 22500
#define PRE_NMS 6000
#define POST_NMS 300
#define MASKW 188          // ceil(6000/32)
#define SORTN 32768
#define CHUNK 32           // K (ci) per LDS stage
#define ASTR 34            // padded LDS row stride (conflict-free, b64-aligned)
#define NCHUNK 144         // 9 taps * 16 ci-chunks

// ---------------------------------------------------------------------------
// Repack conv weights OIHW -> [tap][co][ci] so TDM dim0 runs (ci) are
// contiguous 128B bursts and the DMA writes LDS directly in [co][k] order.
// 9.4MB once per launch, reused 632x from the 192MB L2.
// ---------------------------------------------------------------------------
__global__ void repack_w(const float* __restrict__ Wc, float* __restrict__ Wt)
{
    int i = blockIdx.x * 256 + threadIdx.x;
    if (i >= COUT * CIN * 9) return;
    int ci   = i & 511;
    int rest = i >> 9;
    int co   = rest & 511;
    int tap  = rest >> 9;
    Wt[i] = Wc[(size_t)co * (CIN * 9) + (size_t)ci * 9 + tap];
}

// ---------------------------------------------------------------------------
// 3x3 conv + bias + ReLU as implicit GEMM with V_WMMA_F32_16X16X4_F32.
// grid: (79 m-tiles of 32, 4 co-tiles of 128, 8 batches), block: 256 (8 waves
// as 2M x 4N sub-tiles). Each wave: 16 spatial x 32 cout (two accumulators).
// Weight tiles (128 co x 32 ci) are staged by the Tensor Data Mover into a
// double-buffered LDS region with pad_interval=32DW / pad_amount=2DW, i.e.
// the DMA itself produces the conflict-free 34-float row pitch. One barrier
// per chunk; the next chunk's DMA overlaps the current chunk's 16 WMMAs.
// ---------------------------------------------------------------------------
__global__ __launch_bounds__(256, 1)
void conv3x3_wmma(const float* __restrict__ in, const float* __restrict__ Wt,
                  const float* __restrict__ bc, float* __restrict__ out)
{
    __shared__ float Alds[2][32 * ASTR];      //  ~8.7 KB  [pos][k]
    __shared__ float Blds[2][128 * ASTR];     // ~34.8 KB  [co][k], TDM-filled

    const int tid  = threadIdx.x;
    const int lane = tid & 31;
    const int wave = tid >> 5;
    const int wm   = wave & 1;                // M sub-tile
    const int wn   = wave >> 1;               // N sub-tile (0..3)
    const int m0   = blockIdx.x * 32;
    const int co0  = blockIdx.y * 128;
    const int n    = blockIdx.z;

    v8f acc0 = {};
    v8f acc1 = {};

    const int arow = lane & 15;               // fragment row/col
    const int kgrp = (lane >> 4) * 2;         // K sub-offset per half-wave

    // A loader: thread -> pos = tid&31, k = tid>>5 + 8*j (j=0..3)
    const int lpos = tid & 31;
    const int lkb  = tid >> 5;                // 0..7
    const int m    = m0 + lpos;
    const int y    = m / FW;
    const int x    = m - y * FW;
    const float* inN = in + (size_t)n * CIN * HW;

    // ---- TDM descriptor (CDNA5 ISA): 128 rows x 32 contiguous ci ----
    const unsigned ldsb0 = (unsigned)(size_t)(&Blds[0][0]);
    const unsigned ldsb1 = (unsigned)(size_t)(&Blds[1][0]);
    // group1: data_size=4B | pad_enable | pad_interval=32DW(code 4) |
    //         pad_amount=2DW(code 1); tensor_dim0=32, tensor_dim1=128,
    //         tile_dim0=32, tile_dim1=128, tensor_dim0_stride=512 elements.
    const v8i g1 = { (int)((2u << 16) | (1u << 20) | (4u << 22) | (1u << 25)),
                     (int)(32u << 16),        // tensor_dim0[15:0]
                     (int)(128u << 16),       // tensor_dim0[31:16]=0 | tensor_dim1[15:0]
                     (int)(32u << 16),        // tensor_dim1[31:16]=0 | tile_dim0
                     (int)128,                // tile_dim1 | tile_dim2=0
                     (int)512,                // tensor_dim0_stride[31:0]
                     0, 0 };
    const v4i gz = { 0, 0, 0, 0 };            // 2D tensor: groups 2/3 disabled

    const float* Wbase = Wt + (size_t)co0 * CIN;
    auto issue_tdm = [&](int chunk, unsigned ldsb) {
        int tap = chunk >> 4;
        int cc  = (chunk & 15) << 5;
        unsigned long long ga = (unsigned long long)(size_t)
            (Wbase + (size_t)tap * COUT * CIN + cc);
        v4u g0 = { 1u, ldsb,
                   (unsigned)(ga & 0xFFFFFFFFull),
                   (unsigned)((ga >> 32) & 0x1FFFFFFull) | (2u << 30) };
#if defined(__clang_major__) && (__clang_major__ >= 23)
        v8i g4 = { 0, 0, 0, 0, 0, 0, 0, 0 };
        __builtin_amdgcn_tensor_load_to_lds(g0, g1, gz, gz, g4, 0);
#else
        __builtin_amdgcn_tensor_load_to_lds(g0, g1, gz, gz, 0);
#endif
    };

    if (wave == 0) issue_tdm(0, ldsb0);       // prologue DMA

    int buf = 0;
    for (int chunk = 0; chunk < NCHUNK; ++chunk) {
        const int tap = chunk >> 4;
        const int cc  = (chunk & 15) << 5;
        const int ky = tap / 3, kx = tap - ky * 3;
        const int iy = y + ky - 1, ix = x + kx - 1;
        const bool inb = (m < HW) && (iy >= 0) && (iy < FH) && (ix >= 0) && (ix < FW);
        const float* inTap = inN + iy * FW + ix;

        // stage A: 32 pos x 32 ci (4 elements/thread), other buffer than DMA
        #pragma unroll
        for (int j = 0; j < 4; ++j) {
            int k = lkb + 8 * j;
            Alds[buf][lpos * ASTR + k] = inb ? inTap[(size_t)(cc + k) * HW] : 0.0f;
        }
        if (wave == 0)
            __builtin_amdgcn_s_wait_tensorcnt(0);   // current chunk's DMA done
        __syncthreads();
        // overlap: launch next chunk's weight DMA into the other buffer
        if (wave == 0 && chunk + 1 < NCHUNK)
            issue_tdm(chunk + 1, buf ? ldsb0 : ldsb1);

        const float* Ab = &Alds[buf][0];
        const float* Bb = &Blds[buf][0];
        #pragma unroll
        for (int s = 0; s < 8; ++s) {
            const int kl = s * 4 + kgrp;
            v2f a  = { Ab[(wm * 16 + arow) * ASTR + kl],
                       Ab[(wm * 16 + arow) * ASTR + kl + 1] };
            v2f b0 = { Bb[(wn * 32 + arow) * ASTR + kl],
                       Bb[(wn * 32 + arow) * ASTR + kl + 1] };
            v2f b1 = { Bb[(wn * 32 + 16 + arow) * ASTR + kl],
                       Bb[(wn * 32 + 16 + arow) * ASTR + kl + 1] };
            acc0 = __builtin_amdgcn_wmma_f32_16x16x4_f32(
                       false, a, false, b0, (short)0, acc0, false, false);
            acc1 = __builtin_amdgcn_wmma_f32_16x16x4_f32(
                       false, a, false, b1, (short)0, acc1, false, false);
        }
        buf ^= 1;
    }

    const int coA = co0 + wn * 32 + arow;
    const int coB = coA + 16;
    const float biasA = bc[coA];
    const float biasB = bc[coB];
    float* outA = out + (size_t)n * COUT * HW + (size_t)coA * HW;
    float* outB = out + (size_t)n * COUT * HW + (size_t)coB * HW;
    #pragma unroll
    for (int r = 0; r < 8; ++r) {
        int mm = m0 + wm * 16 + r + 8 * (lane >> 4);
        if (mm < HW) {
            float v0 = acc0[r] + biasA;
            float v1 = acc1[r] + biasB;
            outA[mm] = v0 > 0.0f ? v0 : 0.0f;
            outB[mm] = v1 > 0.0f ? v1 : 0.0f;
        }
    }
}

// ---------------------------------------------------------------------------
// Pack head weights (reg 36ch + cls 18ch -> 64 padded) into workspace.
// ---------------------------------------------------------------------------
__global__ void pack_head(const float* __restrict__ Wcls, const float* __restrict__ bcls,
                          const float* __restrict__ Wreg, const float* __restrict__ breg,
                          float* __restrict__ Wh, float* __restrict__ bh)
{
    int i = blockIdx.x * 256 + threadIdx.x;
    if (i < 64 * 512) {
        int ch = i >> 9, k = i & 511;
        float v = 0.0f;
        if (ch < 36)      v = Wreg[ch * 512 + k];
        else if (ch < 54) v = Wcls[(ch - 36) * 512 + k];
        Wh[i] = v;
    }
    if (i < 64)
        bh[i] = (i < 36) ? breg[i] : (i < 54 ? bcls[i - 36] : 0.0f);
}

// ---------------------------------------------------------------------------
// Fused 1x1 heads: 20000 x 64 x 512 WMMA GEMM, output written in the
// reference's transposed (n, HW*9, C) layout.
// ---------------------------------------------------------------------------
__global__ __launch_bounds__(256, 1)
void head_wmma(const float* __restrict__ x, const float* __restrict__ Wh,
               const float* __restrict__ bh, float* __restrict__ locs,
               float* __restrict__ scores)
{
    __shared__ float Alds[32 * 18];   // [pos][k], padded
    __shared__ float Blds[64 * 18];   // [ch][k], padded

    const int tid  = threadIdx.x;
    const int lane = tid & 31;
    const int wave = tid >> 5;
    const int wm   = wave & 1;
    const int wn   = wave >> 1;
    const int m0   = blockIdx.x * 32;
    const int n    = blockIdx.y;
    const float* xN = x + (size_t)n * CIN * HW;

    v8f acc = {};
    const int kgrp = (lane >> 4) * 2;
    const int ncol = lane & 15;

    for (int cc = 0; cc < CIN; cc += 16) {
        #pragma unroll
        for (int i = 0; i < 2; ++i) {
            int idx = tid + 256 * i;
            int pos = idx & 31, k = idx >> 5;
            int mm  = m0 + pos;
            Alds[pos * 18 + k] = (mm < HW) ? xN[(size_t)(cc + k) * HW + mm] : 0.0f;
        }
        #pragma unroll
        for (int i = 0; i < 4; ++i) {
            int idx = tid + 256 * i;
            int ch = idx >> 4, k = idx & 15;
            Blds[ch * 18 + k] = Wh[ch * 512 + cc + k];
        }
        __syncthreads();
        #pragma unroll
        for (int s = 0; s < 4; ++s) {
            const int kl = s * 4 + kgrp;
            v2f a = { Alds[(wm * 16 + ncol) * 18 + kl],
                      Alds[(wm * 16 + ncol) * 18 + kl + 1] };
            v2f b = { Blds[(wn * 16 + ncol) * 18 + kl],
                      Blds[(wn * 16 + ncol) * 18 + kl + 1] };
            acc = __builtin_amdgcn_wmma_f32_16x16x4_f32(
                      false, a, false, b, (short)0, acc, false, false);
        }
        __syncthreads();
    }

    const int ch = wn * 16 + ncol;
    const float bias = (ch < 54) ? bh[ch] : 0.0f;
    #pragma unroll
    for (int r = 0; r < 8; ++r) {
        int mm = m0 + wm * 16 + r + 8 * (lane >> 4);
        if (mm < HW) {
            float v = acc[r] + bias;
            if (ch < 36)
                locs[(size_t)n * (HW * 36) + (size_t)mm * 36 + ch] = v;
            else if (ch < 54)
                scores[(size_t)n * (HW * 18) + (size_t)mm * 18 + (ch - 36)] = v;
        }
    }
}

// ---------------------------------------------------------------------------
// Anchors
// ---------------------------------------------------------------------------
__global__ void anchors_k(float* __restrict__ anc)
{
    int i = blockIdx.x * 256 + threadIdx.x;
    if (i >= NANCH) return;
    int m = i / 9, a = i - m * 9;
    int gy = m / FW, gx = m - gy * FW;
    float sy = gy * 16.0f, sx = gx * 16.0f;
    const float ratios[3] = {0.5f, 1.0f, 2.0f};
    const float scales[3] = {8.0f, 16.0f, 32.0f};
    float r = ratios[a / 3], s = scales[a % 3];
    float hh = 16.0f * s * sqrtf(r);
    float ww = 16.0f * s * sqrtf(1.0f / r);
    const float c = 8.0f;
    anc[i * 4 + 0] = sy + c - hh * 0.5f;
    anc[i * 4 + 1] = sx + c - ww * 0.5f;
    anc[i * 4 + 2] = sy + c + hh * 0.5f;
    anc[i * 4 + 3] = sx + c + ww * 0.5f;
}

// ---------------------------------------------------------------------------
// Softmax foreground prob (batch 0)
// ---------------------------------------------------------------------------
__global__ void softmax_fg(const float* __restrict__ scores, float* __restrict__ fg)
{
    int i = blockIdx.x * 256 + threadIdx.x;
    if (i >= NANCH) return;
    float s0 = scores[i * 2 + 0], s1 = scores[i * 2 + 1];
    float mx = fmaxf(s0, s1);
    float e0 = expf(s0 - mx), e1 = expf(s1 - mx);
    fg[i] = e1 / (e0 + e1);
}

// ---------------------------------------------------------------------------
// Box decode + clip + min-size filter
// ---------------------------------------------------------------------------
__global__ void decode_k(const float* __restrict__ locs, const float* __restrict__ anc,
                         const float* __restrict__ fg, const int* __restrict__ ih,
                         const int* __restrict__ iw, float* __restrict__ boxes,
                         float* __restrict__ sc)
{
    int i = blockIdx.x * 256 + threadIdx.x;
    if (i >= NANCH) return;
    float a0 = anc[i * 4], a1 = anc[i * 4 + 1], a2 = anc[i * 4 + 2], a3 = anc[i * 4 + 3];
    float ah = a2 - a0, aw = a3 - a1;
    float acy = a0 + 0.5f * ah, acx = a1 + 0.5f * aw;
    float l0 = locs[i * 4], l1 = locs[i * 4 + 1], l2 = locs[i * 4 + 2], l3 = locs[i * 4 + 3];
    float cy = l0 * ah + acy, cx = l1 * aw + acx;
    float h = expf(l2) * ah, w = expf(l3) * aw;
    float Hc = (float)ih[0], Wc = (float)iw[0];
    float y1 = fminf(fmaxf(cy - 0.5f * h, 0.0f), Hc);
    float x1 = fminf(fmaxf(cx - 0.5f * w, 0.0f), Wc);
    float y2 = fminf(fmaxf(cy + 0.5f * h, 0.0f), Hc);
    float x2 = fminf(fmaxf(cx + 0.5f * w, 0.0f), Wc);
    boxes[i * 4 + 0] = y1; boxes[i * 4 + 1] = x1;
    boxes[i * 4 + 2] = y2; boxes[i * 4 + 3] = x2;
    bool valid = ((y2 - y1) >= 16.0f) && ((x2 - x1) >= 16.0f);
    sc[i] = valid ? fg[i] : -INFINITY;
}

// ---------------------------------------------------------------------------
// Bitonic descending sort of (score, idx), N = 32768 padded with -inf
// ---------------------------------------------------------------------------
__global__ void sort_init(const float* __restrict__ sc, float* __restrict__ ss,
                          int* __restrict__ si)
{
    int i = blockIdx.x * 256 + threadIdx.x;
    if (i >= SORTN) return;
    ss[i] = (i < NANCH) ? sc[i] : -INFINITY;
    si[i] = (i < NANCH) ? i : 0;
}

__global__ void bitonic_pass(float* __restrict__ ss, int* __restrict__ si, int j, int k)
{
    int i = blockIdx.x * 256 + threadIdx.x;
    int l = i ^ j;
    if (l > i) {
        bool dirDesc = ((i & k) == 0);
        float a = ss[i], b = ss[l];
        if ((a < b) == dirDesc) {
            ss[i] = b; ss[l] = a;
            int t = si[i]; si[i] = si[l]; si[l] = t;
        }
    }
}

// ---------------------------------------------------------------------------
// Gather top-6000 candidates, build initial keep bits, zero rois
// ---------------------------------------------------------------------------
__global__ void gather_cand(const float* __restrict__ ss, const int* __restrict__ si,
                            const float* __restrict__ boxes, float* __restrict__ cand,
                            unsigned* __restrict__ keepb, float* __restrict__ rois)
{
    int i = blockIdx.x * 256 + threadIdx.x;
    if (i < PRE_NMS) {
        int id = si[i];
        cand[i * 4 + 0] = boxes[id * 4 + 0];
        cand[i * 4 + 1] = boxes[id * 4 + 1];
        cand[i * 4 + 2] = boxes[id * 4 + 2];
        cand[i * 4 + 3] = boxes[id * 4 + 3];
    }
    if (i < MASKW) {
        unsigned bits = 0;
        for (int b = 0; b < 32; ++b) {
            int j = i * 32 + b;
            if (j < PRE_NMS && ss[j] > -1e37f) bits |= (1u << b);
        }
        keepb[i] = bits;
    }
    if (i < POST_NMS * 4) rois[i] = 0.0f;
}

// ---------------------------------------------------------------------------
// NMS suppression bitmask: bit (i,j) set iff j>i and IoU>0.7
// ---------------------------------------------------------------------------
__global__ void nms_mask(const float* __restrict__ cand, unsigned* __restrict__ mask)
{
    int i = blockIdx.x;
    int w = threadIdx.x;
    if (w >= MASKW) return;
    float y1 = cand[i * 4], x1 = cand[i * 4 + 1], y2 = cand[i * 4 + 2], x2 = cand[i * 4 + 3];
    float ai = (y2 - y1) * (x2 - x1);
    unsigned bits = 0;
    for (int b = 0; b < 32; ++b) {
        int j = w * 32 + b;
        if (j < PRE_NMS && j > i) {
            float by1 = cand[j * 4], bx1 = cand[j * 4 + 1];
            float by2 = cand[j * 4 + 2], bx2 = cand[j * 4 + 3];
            float yy1 = fmaxf(y1, by1), xx1 = fmaxf(x1, bx1);
            float yy2 = fminf(y2, by2), xx2 = fminf(x2, bx2);
            float inter = fmaxf(yy2 - yy1, 0.0f) * fmaxf(xx2 - xx1, 0.0f);
            float aj = (by2 - by1) * (bx2 - bx1);
            float iou = inter / (ai + aj - inter + 1e-12f);
            if (iou > 0.7f) bits |= (1u << b);
        }
    }
    mask[(size_t)i * MASKW + w] = bits;
}

// ---------------------------------------------------------------------------
// Greedy NMS sweep + write first 300 kept boxes. Single block.
// ---------------------------------------------------------------------------
__global__ __launch_bounds__(256, 1)
void nms_greedy(const unsigned* __restrict__ mask, const unsigned* __restrict__ keep0,
                const float* __restrict__ cand, float* __restrict__ rois)
{
    __shared__ unsigned keep[MASKW];
    int tid = threadIdx.x;
    for (int w = tid; w < MASKW; w += 256) keep[w] = keep0[w];
    __syncthreads();
    for (int i = 0; i < PRE_NMS; ++i) {
        bool ki = (keep[i >> 5] >> (i & 31)) & 1u;
        __syncthreads();
        if (ki) {
            for (int w = tid; w < MASKW; w += 256)
                keep[w] &= ~mask[(size_t)i * MASKW + w];
        }
        __syncthreads();
    }
    if (tid == 0) {
        int rank = 0;
        for (int i = 0; i < PRE_NMS && rank < POST_NMS; ++i) {
            if ((keep[i >> 5] >> (i & 31)) & 1u) {
                rois[rank * 4 + 0] = cand[i * 4 + 0];
                rois[rank * 4 + 1] = cand[i * 4 + 1];
                rois[rank * 4 + 2] = cand[i * 4 + 2];
                rois[rank * 4 + 3] = cand[i * 4 + 3];
                ++rank;
            }
        }
    }
}

// ---------------------------------------------------------------------------
extern "C" void kernel_launch(void* const* d_in, const int* in_sizes, int n_in,
                              void* d_out, int out_size, void* d_ws, size_t ws_size,
                              hipStream_t stream)
{
    (void)in_sizes; (void)n_in; (void)out_size; (void)ws_size;

    const float* feat  = (const float*)d_in[0];
    const float* Wconv = (const float*)d_in[1];
    const float* bconv = (const float*)d_in[2];
    const float* Wcls  = (const float*)d_in[3];
    const float* bcls  = (const float*)d_in[4];
    const float* Wreg  = (const float*)d_in[5];
    const float* breg  = (const float*)d_in[6];
    const int*   img_h = (const int*)d_in[7];
    const int*   img_w = (const int*)d_in[8];

    // output layout: locs | scores | anchors | rois
    float* out      = (float*)d_out;
    float* locs     = out;                       // 8*22500*4 = 720000
    float* scores   = out + 720000;              // 8*22500*2 = 360000
    float* anc      = out + 1080000;             // 22500*4   = 90000
    float* rois     = out + 1170000;             // 300*4     = 1200

    // workspace layout (floats)
    float* X     = (float*)d_ws;                 // 8*512*2500 = 10,240,000
    float* WT    = X + 10240000;                 // 9*512*512  = 2,359,296
    float* WH    = WT + 2359296;                 // 64*512
    float* BH    = WH + 32768;                   // 64
    float* FG    = BH + 64;                      // 22500
    float* BOX   = FG + 22500;                   // 90000
    float* SCV   = BOX + 90000;                  // 22500
    float* SS    = SCV + 22500;                  // 32768
    int*   SI    = (int*)(SS + 32768);           // 32768
    float* CAND  = (float*)(SI + 32768);         // 24000
    unsigned* KEEPB = (unsigned*)(CAND + 24000); // 192
    unsigned* MASK  = KEEPB + 192;               // 6000*188 = 1,128,000

    // 0. repack weights for TDM-friendly contiguous ci runs
    repack_w<<<(COUT * CIN * 9 + 255) / 256, 256, 0, stream>>>(Wconv, WT);

    // 1. heavy 3x3 conv (WMMA fp32, double-buffered TDM weight staging)
    conv3x3_wmma<<<dim3(79, 4, NB), 256, 0, stream>>>(feat, WT, bconv, X);

    // 2. fused 1x1 heads (WMMA fp32)
    pack_head<<<128, 256, 0, stream>>>(Wcls, bcls, Wreg, breg, WH, BH);
    head_wmma<<<dim3(79, NB), 256, 0, stream>>>(X, WH, BH, locs, scores);

    // 3. anchors
    anchors_k<<<(NANCH + 255) / 256, 256, 0, stream>>>(anc);

    // 4. softmax fg (batch 0) + decode
    softmax_fg<<<(NANCH + 255) / 256, 256, 0, stream>>>(scores, FG);
    decode_k<<<(NANCH + 255) / 256, 256, 0, stream>>>(locs, anc, FG, img_h, img_w, BOX, SCV);

    // 5. descending bitonic sort of 32768 padded (score, idx)
    sort_init<<<SORTN / 256, 256, 0, stream>>>(SCV, SS, SI);
    for (int k = 2; k <= SORTN; k <<= 1)
        for (int j = k >> 1; j > 0; j >>= 1)
            bitonic_pass<<<SORTN / 256, 256, 0, stream>>>(SS, SI, j, k);

    // 6. NMS
    gather_cand<<<(PRE_NMS + 255) / 256, 256, 0, stream>>>(SS, SI, BOX, CAND, KEEPB, rois);
    nms_mask<<<PRE_NMS, 192, 0, stream>>>(CAND, MASK);
    nms_greedy<<<1, 256, 0, stream>>>(MASK, KEEPB, CAND, rois);
}